// PointLoss_1408749273445
// MI455X (gfx1250) — compile-verified
//
#include <hip/hip_runtime.h>
#include <hip/hip_bf16.h>

typedef __attribute__((ext_vector_type(2))) float v2f;
typedef __attribute__((ext_vector_type(8))) float v8f;

#define NPTS 4096
#define NB   16
#define WAVES_PER_BLOCK 8

// One block: 8 waves, each wave owns a 16-row strip of P and sweeps all 4096
// columns of Q in 16-wide tiles. One V_WMMA_F32_16X16X4_F32 per tile yields
// the finished 16x16 distance tile (A=(p,1), B=(-2q,|q|^2), C=|p|^2).
// Running v_min over the accumulator layout + lane butterfly gives row mins.
__global__ __launch_bounds__(256) void chamfer_rowmin_kernel(
    const float* __restrict__ P,   // [NB, NPTS, 3] "row" set
    const float* __restrict__ Q,   // [NB, NPTS, 3] "column" set
    float* __restrict__ rowmin)    // [NB, NPTS]  min over columns
{
    __shared__ float sQ[NPTS * 4];   // per column j: (-2qx, -2qy, -2qz, |q|^2)

    const int b   = blockIdx.y;
    const int tid = threadIdx.x;

    // Stage transformed Q into LDS (64KB, well under the 320KB WGP budget).
    const float* qb = Q + (size_t)b * NPTS * 3;
    for (int j = tid; j < NPTS; j += 256) {
        const float qx = qb[j * 3 + 0];
        const float qy = qb[j * 3 + 1];
        const float qz = qb[j * 3 + 2];
        float4 t;
        t.x = -2.0f * qx;
        t.y = -2.0f * qy;
        t.z = -2.0f * qz;
        t.w = qx * qx + qy * qy + qz * qz;
        *(float4*)&sQ[j * 4] = t;
    }
    __syncthreads();

    const int wave = tid >> 5;
    const int lane = tid & 31;
    const int half = lane >> 4;   // K-half: lanes 0-15 hold K=0,1; 16-31 hold K=2,3
    const int l    = lane & 15;
    const int i0   = (blockIdx.x * WAVES_PER_BLOCK + wave) * 16;

    // A operand: row M=l, K components (half*2, half*2+1); virtual K=3 column = 1.0
    const float* pb = P + (size_t)b * NPTS * 3;
    const float px = pb[(i0 + l) * 3 + 0];
    const float py = pb[(i0 + l) * 3 + 1];
    const float pz = pb[(i0 + l) * 3 + 2];
    const float sq = px * px + py * py + pz * pz;

    v2f a;
    a.x = half ? pz   : px;
    a.y = half ? 1.0f : py;

    // C init: VGPR v holds row (v + half*8) -> broadcast that row's |p|^2.
    v8f c;
#pragma unroll
    for (int v = 0; v < 8; ++v)
        c[v] = __shfl(sq, v + half * 8, 32);

    v8f rmin;
#pragma unroll
    for (int v = 0; v < 8; ++v)
        rmin[v] = 3.402823466e38f;

#pragma unroll 4
    for (int jt = 0; jt < NPTS / 16; ++jt) {
        // B operand: column N=l of this tile, K components (half*2, half*2+1).
        // ds_load_b64, 8B aligned, conflict-free across the half.
        const float2 bq = *(const float2*)&sQ[((jt * 16 + l) << 2) + half * 2];
        v2f bv;
        bv.x = bq.x;
        bv.y = bq.y;

        // D = A x B + C : finished 16x16 squared-distance tile.
        v8f d = __builtin_amdgcn_wmma_f32_16x16x4_f32(
            /*neg_a=*/false, a, /*neg_b=*/false, bv,
            /*c_mod=*/(short)0, c, /*reuse_a=*/false, /*reuse_b=*/false);

#pragma unroll
        for (int v = 0; v < 8; ++v)
            rmin[v] = fminf(rmin[v], d[v]);
    }

    // Reduce over the 16 columns held across lanes of each half (xor < 16
    // stays inside the half).
#pragma unroll
    for (int off = 1; off < 16; off <<= 1) {
#pragma unroll
        for (int v = 0; v < 8; ++v)
            rmin[v] = fminf(rmin[v], __shfl_xor(rmin[v], off, 32));
    }

    if (l == 0) {
#pragma unroll
        for (int v = 0; v < 8; ++v)
            rowmin[(size_t)b * NPTS + i0 + half * 8 + v] = rmin[v];
    }
}

// Sum both rowmin arrays (2*NB*NPTS floats, stored contiguously) and apply
// 100 * 0.5 / (B*N).
__global__ __launch_bounds__(512) void chamfer_reduce_kernel(
    const float* __restrict__ w, float* __restrict__ out)
{
    __shared__ float s[512];
    float acc = 0.0f;
    for (int i = threadIdx.x; i < 2 * NB * NPTS; i += 512)
        acc += w[i];
    s[threadIdx.x] = acc;
    __syncthreads();
#pragma unroll
    for (int stride = 256; stride > 0; stride >>= 1) {
        if ((int)threadIdx.x < stride)
            s[threadIdx.x] += s[threadIdx.x + stride];
        __syncthreads();
    }
    if (threadIdx.x == 0)
        out[0] = s[0] * (50.0f / (float)(NB * NPTS));
}

extern "C" void kernel_launch(void* const* d_in, const int* in_sizes, int n_in,
                              void* d_out, int out_size, void* d_ws, size_t ws_size,
                              hipStream_t stream) {
    (void)in_sizes; (void)n_in; (void)out_size; (void)ws_size;
    const float* a1 = (const float*)d_in[0];   // [16, 4096, 3] f32
    const float* a2 = (const float*)d_in[1];   // [16, 4096, 3] f32
    float* out = (float*)d_out;                // scalar f32

    float* w1 = (float*)d_ws;                  // [16, 4096] min over j (av_dist2 path)
    float* w2 = w1 + (size_t)NB * NPTS;        // [16, 4096] min over i (av_dist1 path)

    dim3 grid(NPTS / (16 * WAVES_PER_BLOCK), NB);   // (32, 16)
    dim3 block(256);

    // Pass 1: rows = array1  -> min over array2 columns.
    chamfer_rowmin_kernel<<<grid, block, 0, stream>>>(a1, a2, w1);
    // Pass 2: swap roles (D is the transpose) -> min over array1 columns.
    chamfer_rowmin_kernel<<<grid, block, 0, stream>>>(a2, a1, w2);

    chamfer_reduce_kernel<<<1, 512, 0, stream>>>(w1, out);
}